// SoftSphereModel_71064528880283
// MI455X (gfx1250) — compile-verified
//
#include <hip/hip_runtime.h>
#include <math.h>

// ---------------------------------------------------------------------------
// Soft-sphere pair potential for MI455X (gfx1250, wave32).
//
// Roofline: 16M pairs * 20B streamed = 320MB @ 23.3 TB/s ~= 14us floor.
// Atom accumulators (500K atoms * 16B = 8MB) are L2-resident (192MB L2);
// scatter adds use native global_atomic_add_f32. Streams use TH=NT loads so
// they don't evict the atom arrays from L2. Scalar outputs (energy, stress)
// reduce per-wave (wave32 shfl_xor) -> per-block (LDS across 8 waves) ->
// per-block partials in d_ws -> final V_WMMA_F32_16X16X4_F32 reduction
// (ones-matrix * partials = column sums), single wave.
// ---------------------------------------------------------------------------

typedef __attribute__((ext_vector_type(2))) float v2f;
typedef __attribute__((ext_vector_type(8))) float v8f;

#define NBLK 1024        // must be a multiple of 4 (WMMA reduce eats 4 rows/step)
#define NTHR 256         // 8 waves per workgroup on CDNA5

__global__ __launch_bounds__(NTHR) void ss_pairs_kernel(
    const float* __restrict__ dr_vec,   // [E,3]
    const int*   __restrict__ map0,     // [E]
    const int*   __restrict__ map1,     // [E]
    const float* __restrict__ cell,     // [3,3]
    float* __restrict__ atom_e,         // [n]
    float* __restrict__ forces,         // [n,3]
    float* __restrict__ partials,       // [NBLK,16] in d_ws
    int npairs)
{
    float e_acc = 0.0f;
    float s00 = 0.f, s01 = 0.f, s02 = 0.f;
    float s10 = 0.f, s11 = 0.f, s12 = 0.f;
    float s20 = 0.f, s21 = 0.f, s22 = 0.f;

    const int stride = gridDim.x * blockDim.x;
    for (int e = blockIdx.x * blockDim.x + threadIdx.x; e < npairs; e += stride) {
        // Non-temporal streaming loads: keep L2 free for the atom accumulators.
        float dx = __builtin_nontemporal_load(&dr_vec[3 * e + 0]);
        float dy = __builtin_nontemporal_load(&dr_vec[3 * e + 1]);
        float dz = __builtin_nontemporal_load(&dr_vec[3 * e + 2]);
        int   i  = __builtin_nontemporal_load(&map0[e]);
        int   j  = __builtin_nontemporal_load(&map1[e]);

        float r2 = dx * dx + dy * dy + dz * dz;
        float r  = sqrtf(r2);
        float x  = 1.0f - r;                      // 1 - dr/sigma  (sigma = 1)
        bool  in = r < 1.0f;

        // eps/alpha * x^2 = 0.5*x^2 ; half-share per endpoint -> 0.25*x^2
        float ep_h = in ? 0.25f * x * x : 0.0f;
        // pair_force / r = (-eps/sigma * x) / r = -x / r   (zero outside cutoff)
        float s = in ? (-x / r) : 0.0f;
        float fx = s * dx, fy = s * dy, fz = s * dz;

        // 8 scatter-adds per pair; all land in L2 (8MB working set).
        unsafeAtomicAdd(&atom_e[i], ep_h);
        unsafeAtomicAdd(&atom_e[j], ep_h);
        unsafeAtomicAdd(&forces[3 * i + 0],  fx);
        unsafeAtomicAdd(&forces[3 * i + 1],  fy);
        unsafeAtomicAdd(&forces[3 * i + 2],  fz);
        unsafeAtomicAdd(&forces[3 * j + 0], -fx);
        unsafeAtomicAdd(&forces[3 * j + 1], -fy);
        unsafeAtomicAdd(&forces[3 * j + 2], -fz);

        e_acc += ep_h;                 // energy = 0.5 * sum(pair_e) = sum(ep_h)
        s00 += dx * fx; s01 += dx * fy; s02 += dx * fz;
        s10 += dy * fx; s11 += dy * fy; s12 += dy * fz;
        s20 += dz * fx; s21 += dz * fy; s22 += dz * fz;
    }

    // ---- wave32 butterfly reduction of the 10 scalars -----------------------
    float acc[10] = {e_acc, s00, s01, s02, s10, s11, s12, s20, s21, s22};
#pragma unroll
    for (int c = 0; c < 10; ++c) {
#pragma unroll
        for (int m = 16; m >= 1; m >>= 1)
            acc[c] += __shfl_xor(acc[c], m, 32);
    }

    // ---- combine 8 waves of the block through LDS ---------------------------
    __shared__ float red[NTHR / 32][10];
    const int lane = threadIdx.x & 31;
    const int wave = threadIdx.x >> 5;
    if (lane == 0) {
#pragma unroll
        for (int c = 0; c < 10; ++c) red[wave][c] = acc[c];
    }
    __syncthreads();

    if (threadIdx.x < 16) {
        float t = 0.0f;
        if (threadIdx.x < 10) {
#pragma unroll
            for (int w = 0; w < NTHR / 32; ++w) t += red[w][threadIdx.x];
            if (threadIdx.x >= 1) {
                // stress = -sum(outer(dr, fvec)) / |det(cell)|; pre-scale here
                float det = cell[0] * (cell[4] * cell[8] - cell[5] * cell[7])
                          - cell[1] * (cell[3] * cell[8] - cell[5] * cell[6])
                          + cell[2] * (cell[3] * cell[7] - cell[4] * cell[6]);
                t *= -1.0f / fabsf(det);
            }
        }
        partials[blockIdx.x * 16 + threadIdx.x] = t;   // pads [10..15] with 0
    }
}

// ---------------------------------------------------------------------------
// Final reduction: sum NBLK rows of 16 partials with V_WMMA_F32_16X16X4_F32.
// A = all-ones 16x4, B = 4 consecutive partial rows (K) x 16 components (N);
// C[m][n] accumulates sum_k B[k][n] for every m. With all-ones A the result
// is invariant to the exact K striping of the B operand. Single wave; EXEC
// is all-1s through every WMMA (divergence only after the loop).
// ---------------------------------------------------------------------------
__global__ __launch_bounds__(32) void ss_reduce_wmma_kernel(
    const float* __restrict__ partials,   // [NBLK,16]
    float* __restrict__ out,              // d_out base
    int stress_base)                      // 1 + 4*n_atoms
{
    const int lane = threadIdx.x;
    const int n  = lane & 15;        // B/C column
    const int kh = lane >> 4;        // K half selector for the 2 B VGPRs

    v2f a; a.x = 1.0f; a.y = 1.0f;   // all-ones A matrix (16x4 striped)
    v8f c = {};

    for (int b = 0; b < NBLK; b += 4) {
        v2f bm;
        bm.x = partials[(b + 2 * kh + 0) * 16 + n];
        bm.y = partials[(b + 2 * kh + 1) * 16 + n];
        c = __builtin_amdgcn_wmma_f32_16x16x4_f32(
                /*neg_a=*/false, a, /*neg_b=*/false, bm,
                /*c_mod=*/(short)0, c, /*reuse_a=*/false, /*reuse_b=*/false);
    }

    // Row M=0 lives in c[0] on lanes 0..15 (N = lane); all rows are equal.
    if (lane < 10) {
        float v = c[0];
        out[lane == 0 ? 0 : (stress_base + lane - 1)] = v;
    }
}

extern "C" void kernel_launch(void* const* d_in, const int* in_sizes, int n_in,
                              void* d_out, int out_size, void* d_ws, size_t ws_size,
                              hipStream_t stream) {
    const float* dr_vec  = (const float*)d_in[0];
    const int*   mapping = (const int*)d_in[1];
    const float* cell    = (const float*)d_in[2];
    // d_in[3] = n_atoms (device scalar) — recover host-side from out_size:
    // out_size = 1 + n + 3n + 9  =>  n = (out_size - 10) / 4
    const int n_atoms = (out_size - 10) / 4;
    const int npairs  = in_sizes[0] / 3;

    const int* map0 = mapping;
    const int* map1 = mapping + npairs;

    float* out    = (float*)d_out;
    float* atom_e = out + 1;
    float* forces = out + 1 + n_atoms;
    const int stress_base = 1 + 4 * n_atoms;

    float* partials = (float*)d_ws;        // NBLK*16 floats = 64 KB

    // Outputs are accumulated via atomics -> zero them every call (capturable).
    hipMemsetAsync(d_out, 0, (size_t)out_size * sizeof(float), stream);

    ss_pairs_kernel<<<NBLK, NTHR, 0, stream>>>(
        dr_vec, map0, map1, cell, atom_e, forces, partials, npairs);

    ss_reduce_wmma_kernel<<<1, 32, 0, stream>>>(partials, out, stress_base);
}